// InteractionPPBlockSWM_32384053412126
// MI455X (gfx1250) — compile-verified
//
#include <hip/hip_runtime.h>

#define HIDDEN 128
#define IEMB   64
#define SBF_K  42
#define SBF_KP 44
#define RAD    6

typedef __attribute__((ext_vector_type(2))) float v2f;
typedef __attribute__((ext_vector_type(8))) float v8f;

__device__ __forceinline__ float silu_f(float v) {
    return v * (1.0f / (1.0f + __expf(-v)));
}

// Wave-level GEMM: NT row-subtiles of 16 x one 16-col slab, K in steps of 4,
// via V_WMMA_F32_16X16X4_F32.  A comes from LDS (rows m = lane&15, k-pair
// 2*(lane>>4)+{0,1} -> ds_load_2addr_b64).  B comes pre-packed in global:
// Bp[((k>>2)*N + n)*4 + (k&3)] so each lane's two elements are one aligned
// b64 load, reused across all NT row-subtiles.
template <int NT>
__device__ __forceinline__ void wmma_gemm_packed(const float* A, int lda,
                                                 const float* __restrict__ Bp, int N,
                                                 int n0, int K, v8f* c)
{
    const int lane = threadIdx.x & 31;
    const int m    = lane & 15;
    const int kh   = (lane >> 4) << 1;    // 0 or 2
    const int n    = n0 + (lane & 15);
    const float* bp = Bp + (size_t)n * 4 + kh;
    for (int k4 = 0; k4 < (K >> 2); ++k4) {
        v2f b = *(const v2f*)(bp + (size_t)k4 * N * 4);
        #pragma unroll
        for (int rt = 0; rt < NT; ++rt) {
            const float* ar = A + (size_t)(rt * 16 + m) * lda + (k4 << 2) + kh;
            v2f a;
            a.x = ar[0];
            a.y = ar[1];
            c[rt] = __builtin_amdgcn_wmma_f32_16x16x4_f32(false, a, false, b,
                                                          (short)0, c[rt],
                                                          false, false);
        }
    }
}

// Repack a KxN row-major weight into the fragment-native layout above.
__global__ void pack_weight_kernel(const float* __restrict__ W,
                                   float* __restrict__ Wp, int K, int N)
{
    int i = blockIdx.x * blockDim.x + threadIdx.x;
    if (i < K * N) {
        int k = i / N, n = i % N;
        Wp[((size_t)(k >> 2) * N + n) * 4 + (k & 3)] = W[i];
    }
}

// Fold basis MLPs: WcR[6][128] = W_rbf1@W_rbf2 (plain, used in VALU epilogue),
// WcS = W_sbf1@W_sbf2 written directly in packed layout, K padded 42 -> 44.
__global__ void precompute_kernel(const float* __restrict__ Wr1, const float* __restrict__ Wr2,
                                  const float* __restrict__ Ws1, const float* __restrict__ Ws2,
                                  float* __restrict__ WcR, float* __restrict__ WcSp)
{
    int tid = blockIdx.x * blockDim.x + threadIdx.x;
    int stride = blockDim.x * gridDim.x;
    for (int i = tid; i < RAD * HIDDEN; i += stride) {
        int r = i / HIDDEN, c = i % HIDDEN;
        float s = 0.f;
        #pragma unroll
        for (int k = 0; k < 8; ++k) s += Wr1[r * 8 + k] * Wr2[k * HIDDEN + c];
        WcR[i] = s;
    }
    for (int i = tid; i < SBF_KP * IEMB; i += stride) {
        int r = i / IEMB, c = i % IEMB;
        float s = 0.f;
        if (r < SBF_K) {
            #pragma unroll
            for (int k = 0; k < 8; ++k) s += Ws1[r * 8 + k] * Ws2[k * IEMB + c];
        }
        WcSp[((size_t)(r >> 2) * IEMB + c) * 4 + (r & 3)] = s;
    }
}

// Stage 1 per 64-edge tile (256 thr = 8 waves, wave -> 16-col slab, 4 row tiles):
//   x_ji = silu(x@Wji+b), x_kj = silu(x@Wkj+b)*rbf_e, xd = silu(x_kj@Wdown)
__global__ __launch_bounds__(256)
void edge_stage1_kernel(const float* __restrict__ x, const float* __restrict__ rbf,
                        const float* __restrict__ WjiP, const float* __restrict__ bji,
                        const float* __restrict__ WkjP, const float* __restrict__ bkj,
                        const float* __restrict__ WcR, const float* __restrict__ WdnP,
                        float* __restrict__ xji_out, float* __restrict__ xd_out, int E)
{
    __shared__ float xs[64][HIDDEN + 4];
    __shared__ float ks[64][HIDDEN + 4];
    __shared__ float rs[64][8];
    const int e0  = blockIdx.x * 64;
    const int tid = threadIdx.x;

    for (int i = tid; i < 64 * (HIDDEN / 4); i += 256) {
        int r = i >> 5, c4 = i & 31;
        int e = min(e0 + r, E - 1);
        float4 v = ((const float4*)(x + (size_t)e * HIDDEN))[c4];
        *(float4*)&xs[r][c4 * 4] = v;
    }
    for (int i = tid; i < 64 * RAD; i += 256) {
        int r = i / RAD, c = i % RAD;
        int e = min(e0 + r, E - 1);
        rs[r][c] = rbf[(size_t)e * RAD + c];
    }
    __syncthreads();

    const int wave = tid >> 5, lane = tid & 31;
    const int n0   = wave * 16;
    const int col  = n0 + (lane & 15);
    const int half = lane >> 4;

    v8f cji[4] = {}; v8f ckj[4] = {};
    wmma_gemm_packed<4>(&xs[0][0], HIDDEN + 4, WjiP, HIDDEN, n0, HIDDEN, cji);
    wmma_gemm_packed<4>(&xs[0][0], HIDDEN + 4, WkjP, HIDDEN, n0, HIDDEN, ckj);

    float wr[RAD];
    #pragma unroll
    for (int j = 0; j < RAD; ++j) wr[j] = WcR[j * HIDDEN + col];
    const float bj = bji[col], bk = bkj[col];

    #pragma unroll
    for (int rt = 0; rt < 4; ++rt) {
        #pragma unroll
        for (int i = 0; i < 8; ++i) {
            int r = rt * 16 + i + 8 * half;   // C layout: vgpr i -> row i+8*(lane>=16)
            float re = 0.f;
            #pragma unroll
            for (int j = 0; j < RAD; ++j) re += rs[r][j] * wr[j];
            ks[r][col] = silu_f(ckj[rt][i] + bk) * re;
            int e = e0 + r;
            if (e < E) xji_out[(size_t)e * HIDDEN + col] = silu_f(cji[rt][i] + bj);
        }
    }
    __syncthreads();

    // down-projection 128 -> 64: 8 waves = 4 col-tiles x 2 row-pair groups
    {
        const int n0d   = (wave & 3) * 16;
        const int rbase = (wave >> 2) * 32;
        const int cold  = n0d + (lane & 15);
        v8f cd[2] = {};
        wmma_gemm_packed<2>(&ks[rbase][0], HIDDEN + 4, WdnP, IEMB, n0d, HIDDEN, cd);
        #pragma unroll
        for (int rt = 0; rt < 2; ++rt) {
            #pragma unroll
            for (int i = 0; i < 8; ++i) {
                int r = rbase + rt * 16 + i + 8 * half;
                int e = e0 + r;
                if (e < E) xd_out[(size_t)e * IEMB + cold] = silu_f(cd[rt][i]);
            }
        }
    }
}

// Triplet stage per 32-triplet tile (128 thr = 4 waves, wave -> 16-col slab,
// 2 row tiles).  sbf_e = sbf @ WcS (WMMA M32 N64 K44), then fused scaled
// gather-scatter:
//   acc[idx_ji] += (alpha + (1-alpha)*lambda[bt[idx_kj]]) * xd[idx_kj] * sbf_e
// (base + weighted segment-sums in ONE atomic pass; acc[E,64]=128MB is L2-resident)
__global__ __launch_bounds__(128)
void triplet_kernel(const float* __restrict__ sbf,
                    const int* __restrict__ idx_kj, const int* __restrict__ idx_ji,
                    const int* __restrict__ bt, const float* __restrict__ lambda_d,
                    const float* __restrict__ alpha_p,
                    const float* __restrict__ WcSp, const float* __restrict__ xd,
                    float* __restrict__ acc, int T)
{
    __shared__ float ss[32][SBF_KP];
    __shared__ int   skj[32], sji[32];
    __shared__ float ssc[32];
    const int t0  = blockIdx.x * 32;
    const int tid = threadIdx.x;

    for (int i = tid; i < 32 * SBF_KP; i += 128) {
        int r = i / SBF_KP, c = i % SBF_KP;
        int t = min(t0 + r, T - 1);
        ss[r][c] = (c < SBF_K) ? sbf[(size_t)t * SBF_K + c] : 0.f;
    }
    if (tid < 32) {
        int t  = min(t0 + tid, T - 1);
        int kj = idx_kj[t];
        skj[tid] = kj;
        sji[tid] = idx_ji[t];
        float al = alpha_p[0];
        ssc[tid] = al + (1.f - al) * lambda_d[bt[kj]];
    }
    __syncthreads();

    const int wave = tid >> 5, lane = tid & 31;
    const int n0   = wave * 16;
    const int col  = n0 + (lane & 15);
    const int half = lane >> 4;

    v8f c[2] = {};
    wmma_gemm_packed<2>(&ss[0][0], SBF_KP, WcSp, IEMB, n0, SBF_KP, c);

    #pragma unroll
    for (int rt = 0; rt < 2; ++rt) {
        #pragma unroll
        for (int i = 0; i < 8; ++i) {
            int r = rt * 16 + i + 8 * half;
            int t = t0 + r;
            if (t < T) {
                float v = c[rt][i] * xd[(size_t)skj[r] * IEMB + col] * ssc[r];
                atomicAdd(&acc[(size_t)sji[r] * IEMB + col], v);
            }
        }
    }
}

// Stage 2 per 64-edge tile: up-proj + skip, residual-before, lin(+x skip),
// residual-after.  Six WMMA slabs ping-ponging between two LDS tiles.
__global__ __launch_bounds__(256)
void edge_stage2_kernel(const float* __restrict__ x,
                        const float* __restrict__ xji, const float* __restrict__ acc,
                        const float* __restrict__ WupP,
                        const float* __restrict__ Wb1P, const float* __restrict__ bb1,
                        const float* __restrict__ Wb2P, const float* __restrict__ bb2,
                        const float* __restrict__ WlP,  const float* __restrict__ bl,
                        const float* __restrict__ Wa1P, const float* __restrict__ ba1,
                        const float* __restrict__ Wa2P, const float* __restrict__ ba2,
                        float* __restrict__ out, int E)
{
    __shared__ float h0[64][HIDDEN + 4];
    __shared__ float h1[64][HIDDEN + 4];
    __shared__ float as[64][IEMB + 4];
    const int e0  = blockIdx.x * 64;
    const int tid = threadIdx.x;

    for (int i = tid; i < 64 * (IEMB / 4); i += 256) {
        int r = i >> 4, c4 = i & 15;
        int e = min(e0 + r, E - 1);
        float4 v = ((const float4*)(acc + (size_t)e * IEMB))[c4];
        *(float4*)&as[r][c4 * 4] = v;
    }
    __syncthreads();

    const int wave = tid >> 5, lane = tid & 31;
    const int n0   = wave * 16;
    const int col  = n0 + (lane & 15);
    const int half = lane >> 4;

    {   // h = x_ji + silu(acc @ W_up)
        v8f c[4] = {};
        wmma_gemm_packed<4>(&as[0][0], IEMB + 4, WupP, HIDDEN, n0, IEMB, c);
        #pragma unroll
        for (int rt = 0; rt < 4; ++rt)
            #pragma unroll
            for (int i = 0; i < 8; ++i) {
                int r = rt * 16 + i + 8 * half;
                int e = min(e0 + r, E - 1);
                h0[r][col] = xji[(size_t)e * HIDDEN + col] + silu_f(c[rt][i]);
            }
    }
    __syncthreads();
    {   // t = silu(h @ Wb1 + bb1)
        v8f c[4] = {};
        wmma_gemm_packed<4>(&h0[0][0], HIDDEN + 4, Wb1P, HIDDEN, n0, HIDDEN, c);
        float b = bb1[col];
        #pragma unroll
        for (int rt = 0; rt < 4; ++rt)
            #pragma unroll
            for (int i = 0; i < 8; ++i) {
                int r = rt * 16 + i + 8 * half;
                h1[r][col] = silu_f(c[rt][i] + b);
            }
    }
    __syncthreads();
    {   // h += silu(t @ Wb2 + bb2)
        v8f c[4] = {};
        wmma_gemm_packed<4>(&h1[0][0], HIDDEN + 4, Wb2P, HIDDEN, n0, HIDDEN, c);
        float b = bb2[col];
        #pragma unroll
        for (int rt = 0; rt < 4; ++rt)
            #pragma unroll
            for (int i = 0; i < 8; ++i) {
                int r = rt * 16 + i + 8 * half;
                h0[r][col] += silu_f(c[rt][i] + b);
            }
    }
    __syncthreads();
    {   // h = silu(h @ Wl + bl) + x
        v8f c[4] = {};
        wmma_gemm_packed<4>(&h0[0][0], HIDDEN + 4, WlP, HIDDEN, n0, HIDDEN, c);
        float b = bl[col];
        #pragma unroll
        for (int rt = 0; rt < 4; ++rt)
            #pragma unroll
            for (int i = 0; i < 8; ++i) {
                int r = rt * 16 + i + 8 * half;
                int e = min(e0 + r, E - 1);
                h1[r][col] = silu_f(c[rt][i] + b) + x[(size_t)e * HIDDEN + col];
            }
    }
    __syncthreads();
    {   // t = silu(h @ Wa1 + ba1)
        v8f c[4] = {};
        wmma_gemm_packed<4>(&h1[0][0], HIDDEN + 4, Wa1P, HIDDEN, n0, HIDDEN, c);
        float b = ba1[col];
        #pragma unroll
        for (int rt = 0; rt < 4; ++rt)
            #pragma unroll
            for (int i = 0; i < 8; ++i) {
                int r = rt * 16 + i + 8 * half;
                h0[r][col] = silu_f(c[rt][i] + b);
            }
    }
    __syncthreads();
    {   // out = h + silu(t @ Wa2 + ba2)
        v8f c[4] = {};
        wmma_gemm_packed<4>(&h0[0][0], HIDDEN + 4, Wa2P, HIDDEN, n0, HIDDEN, c);
        float b = ba2[col];
        #pragma unroll
        for (int rt = 0; rt < 4; ++rt)
            #pragma unroll
            for (int i = 0; i < 8; ++i) {
                int r = rt * 16 + i + 8 * half;
                int e = e0 + r;
                if (e < E) out[(size_t)e * HIDDEN + col] = h1[r][col] + silu_f(c[rt][i] + b);
            }
    }
}

extern "C" void kernel_launch(void* const* d_in, const int* in_sizes, int n_in,
                              void* d_out, int out_size, void* d_ws, size_t ws_size,
                              hipStream_t stream)
{
    (void)n_in; (void)out_size; (void)ws_size;
    // setup_inputs() order, params dict flattened in declaration order.
    const float* x    = (const float*)d_in[0];
    const float* rbf  = (const float*)d_in[1];
    const float* sbf  = (const float*)d_in[2];
    const int*   ikj  = (const int*)  d_in[3];
    const int*   iji  = (const int*)  d_in[4];
    const int*   bt   = (const int*)  d_in[5];
    const float* lam  = (const float*)d_in[6];
    const float* alph = (const float*)d_in[7];
    const float* Wr1  = (const float*)d_in[8];
    const float* Wr2  = (const float*)d_in[9];
    const float* Ws1  = (const float*)d_in[10];
    const float* Ws2  = (const float*)d_in[11];
    const float* Wkj  = (const float*)d_in[12];
    const float* bkj  = (const float*)d_in[13];
    const float* Wji  = (const float*)d_in[14];
    const float* bji  = (const float*)d_in[15];
    const float* Wdn  = (const float*)d_in[16];
    const float* Wup  = (const float*)d_in[17];
    const float* Wb1  = (const float*)d_in[18];
    const float* bb1  = (const float*)d_in[19];
    const float* Wb2  = (const float*)d_in[20];
    const float* bb2  = (const float*)d_in[21];
    const float* Wl   = (const float*)d_in[22];
    const float* bl   = (const float*)d_in[23];
    const float* Wa1  = (const float*)d_in[24];
    const float* ba1  = (const float*)d_in[25];
    const float* Wa2  = (const float*)d_in[26];
    const float* ba2  = (const float*)d_in[27];

    const int E = in_sizes[0] / HIDDEN;
    const int T = in_sizes[3];

    float* ws   = (float*)d_ws;
    float* WcR  = ws;                         // [6][128]
    float* WcSp = WcR  + RAD * HIDDEN;        // packed [44][64]
    float* WjiP = WcSp + SBF_KP * IEMB;       // packed 128x128
    float* WkjP = WjiP + HIDDEN * HIDDEN;
    float* WdnP = WkjP + HIDDEN * HIDDEN;     // packed 128x64
    float* WupP = WdnP + HIDDEN * IEMB;       // packed 64x128
    float* Wb1P = WupP + IEMB * HIDDEN;
    float* Wb2P = Wb1P + HIDDEN * HIDDEN;
    float* WlP  = Wb2P + HIDDEN * HIDDEN;
    float* Wa1P = WlP  + HIDDEN * HIDDEN;
    float* Wa2P = Wa1P + HIDDEN * HIDDEN;
    float* xji  = Wa2P + HIDDEN * HIDDEN;     // [E][128]
    float* xd   = xji  + (size_t)E * HIDDEN;  // [E][64]
    float* acc  = xd   + (size_t)E * IEMB;    // [E][64]

    hipMemsetAsync(acc, 0, (size_t)E * IEMB * sizeof(float), stream);

    precompute_kernel<<<8, 256, 0, stream>>>(Wr1, Wr2, Ws1, Ws2, WcR, WcSp);

    const int pb128 = (HIDDEN * HIDDEN + 255) / 256;
    const int pb64  = (HIDDEN * IEMB  + 255) / 256;
    pack_weight_kernel<<<pb128, 256, 0, stream>>>(Wji, WjiP, HIDDEN, HIDDEN);
    pack_weight_kernel<<<pb128, 256, 0, stream>>>(Wkj, WkjP, HIDDEN, HIDDEN);
    pack_weight_kernel<<<pb64,  256, 0, stream>>>(Wdn, WdnP, HIDDEN, IEMB);
    pack_weight_kernel<<<pb64,  256, 0, stream>>>(Wup, WupP, IEMB, HIDDEN);
    pack_weight_kernel<<<pb128, 256, 0, stream>>>(Wb1, Wb1P, HIDDEN, HIDDEN);
    pack_weight_kernel<<<pb128, 256, 0, stream>>>(Wb2, Wb2P, HIDDEN, HIDDEN);
    pack_weight_kernel<<<pb128, 256, 0, stream>>>(Wl,  WlP,  HIDDEN, HIDDEN);
    pack_weight_kernel<<<pb128, 256, 0, stream>>>(Wa1, Wa1P, HIDDEN, HIDDEN);
    pack_weight_kernel<<<pb128, 256, 0, stream>>>(Wa2, Wa2P, HIDDEN, HIDDEN);

    const int ebl = (E + 63) / 64;
    edge_stage1_kernel<<<ebl, 256, 0, stream>>>(x, rbf, WjiP, bji, WkjP, bkj,
                                                WcR, WdnP, xji, xd, E);

    const int tbl = (T + 31) / 32;
    triplet_kernel<<<tbl, 128, 0, stream>>>(sbf, ikj, iji, bt, lam, alph,
                                            WcSp, xd, acc, T);

    edge_stage2_kernel<<<ebl, 256, 0, stream>>>(x, xji, acc, WupP,
                                                Wb1P, bb1, Wb2P, bb2, WlP, bl,
                                                Wa1P, ba1, Wa2P, ba2,
                                                (float*)d_out, E);
}